// SAGEConv_DA_8040178778268
// MI455X (gfx1250) — compile-verified
//
#include <hip/hip_runtime.h>
#include <hip/hip_bf16.h>

// Problem constants (from reference)
#define N_SRC 50000
#define N_DST 10000
#define KNBR  32
#define IN_F  256
#define OUT_F 128
#define NHEAD 8
#define LDIM  10
#define ATTN_COLS (OUT_F * NHEAD)   // 1024

typedef __attribute__((ext_vector_type(2))) float v2f;
typedef __attribute__((ext_vector_type(8))) float v8f;

// ---------------------------------------------------------------------------
// Kernel 1: C[M x 128] = A[M x 256] @ W[256 x 128] + bias[128]
// fp32 WMMA 16x16x4. Block = 256 threads = 8 wave32; 256x128 output tile
// (32 M-rows per wave -> each B fragment feeds 2 WMMAs). W staged in LDS in
// (even-K, odd-K)-paired layout so a B fragment pair is one ds_load_2addr_b64;
// all 8 B fragments of a k-step are register-batched before the WMMA burst so
// the LDS waits are consolidated instead of one stall per fragment pair.
// ---------------------------------------------------------------------------
#define KCHUNK 64

__global__ __launch_bounds__(256) void gemm_k256_n128_bias(
    const float* __restrict__ A, const float* __restrict__ W,
    const float* __restrict__ bias, float* __restrict__ C, int M)
{
    __shared__ float Wlds[KCHUNK * OUT_F];   // 64*128 floats = 32 KB, paired layout

    const int tid   = threadIdx.x;
    const int wave  = tid >> 5;
    const int lane  = tid & 31;
    const int half  = lane >> 4;   // 0: lanes 0-15, 1: lanes 16-31
    const int mlane = lane & 15;

    const int mbase = blockIdx.x * 256 + wave * 32;
    int r0 = mbase + mlane;                    // A-frag 0 rows (M = lane%16)
    int r1 = mbase + 16 + mlane;               // A-frag 1 rows
    int a0r = (r0 < M) ? r0 : (M - 1);         // clamp: keep EXEC all-ones
    int a1r = (r1 < M) ? r1 : (M - 1);
    const float* A0 = A + (size_t)a0r * IN_F + 2 * half;
    const float* A1 = A + (size_t)a1r * IN_F + 2 * half;

    v8f acc0[8], acc1[8];
#pragma unroll
    for (int t = 0; t < 8; ++t) { acc0[t] = (v8f)0.0f; acc1[t] = (v8f)0.0f; }

    for (int kc = 0; kc < IN_F; kc += KCHUNK) {
        // cooperative load of W chunk into paired LDS layout:
        // W[kc+kk][n] -> Wlds[(kk>>1)*256 + n*2 + (kk&1)]
        for (int idx = tid; idx < KCHUNK * OUT_F; idx += 256) {
            int kk = idx >> 7;            // /128
            int n  = idx & 127;
            Wlds[(kk >> 1) * (2 * OUT_F) + n * 2 + (kk & 1)] =
                W[(size_t)(kc + kk) * OUT_F + n];
        }
        __syncthreads();

#pragma unroll
        for (int k0 = 0; k0 < KCHUNK; k0 += 4) {
            // A fragments: contiguous 8-byte global loads
            v2f a0 = *(const v2f*)(A0 + kc + k0);
            v2f a1 = *(const v2f*)(A1 + kc + k0);
            // Batch all 8 B fragments (4x ds_load_2addr_b64) before WMMAs
            v2f b[8];
#pragma unroll
            for (int t = 0; t < 8; ++t)
                b[t] = *(const v2f*)&Wlds[((k0 >> 1) + half) * (2 * OUT_F) +
                                          (t * 16 + mlane) * 2];
#pragma unroll
            for (int t = 0; t < 8; ++t) {
                acc0[t] = __builtin_amdgcn_wmma_f32_16x16x4_f32(
                    false, a0, false, b[t], (short)0, acc0[t], false, false);
                acc1[t] = __builtin_amdgcn_wmma_f32_16x16x4_f32(
                    false, a1, false, b[t], (short)0, acc1[t], false, false);
            }
        }
        __syncthreads();
    }

    // D layout: VGPR v -> M = v + 8*half (within each 16-row frag), N = mlane+16t
#pragma unroll
    for (int t = 0; t < 8; ++t) {
        int ncol = t * 16 + mlane;
        float bv = bias[ncol];
#pragma unroll
        for (int v = 0; v < 8; ++v) {
            int m0 = mbase + v + 8 * half;
            int m1 = mbase + 16 + v + 8 * half;
            if (m0 < M) C[(size_t)m0 * OUT_F + ncol] = acc0[t][v] + bv;
            if (m1 < M) C[(size_t)m1 * OUT_F + ncol] = acc1[t][v] + bv;
        }
    }
}

// ---------------------------------------------------------------------------
// Kernel 2: collapse the attention projection.
//   v[i]   = sum_j W_attn[i][j] * fa[j]   (i in 0..255)
//   vbuf[256] = sum_j b_attn[j] * fa[j]
// ---------------------------------------------------------------------------
__global__ __launch_bounds__(256) void prep_v_kernel(
    const float* __restrict__ W_attn, const float* __restrict__ b_attn,
    const float* __restrict__ fa, float* __restrict__ vbuf)
{
    int i = threadIdx.x;   // 256 threads, one row each
    float acc = 0.0f;
    for (int j = 0; j < ATTN_COLS; ++j)
        acc = fmaf(W_attn[(size_t)i * ATTN_COLS + j], fa[j], acc);
    vbuf[i] = acc;
    if (i == 0) {
        float c = 0.0f;
        for (int j = 0; j < ATTN_COLS; ++j) c = fmaf(b_attn[j], fa[j], c);
        vbuf[256] = c;
    }
}

// ---------------------------------------------------------------------------
// Kernel 3: per-node attention + aggregation + epilogue. One block per node.
// ---------------------------------------------------------------------------
__global__ __launch_bounds__(256) void attn_aggregate_kernel(
    const float* __restrict__ h_src, const float* __restrict__ self_out,
    const float* __restrict__ labels, const float* __restrict__ label_rela,
    const float* __restrict__ vbuf, const float* __restrict__ eps_p,
    const float* __restrict__ out_bias, const int* __restrict__ nbr_idx,
    float* __restrict__ rst)
{
    __shared__ float hrows[KNBR * OUT_F];   // 32*128 floats = 16 KB
    __shared__ float score[KNBR];
    __shared__ float rsh[16];
    __shared__ float s2sh;

    const int n    = blockIdx.x;
    const int tid  = threadIdx.x;
    const int wave = tid >> 5;
    const int lane = tid & 31;
    const float eps = eps_p[0];

    // r[a] = sum_b label_rela[a][b] * cent_l[n][b]
    if (tid < LDIM) {
        float acc = 0.0f;
        for (int b = 0; b < LDIM; ++b)
            acc = fmaf(label_rela[tid * LDIM + b], labels[(size_t)n * LDIM + b], acc);
        rsh[tid] = acc;
    }
    // s2 = dot(cent_f[n], v2) + c   (wave 7)
    if (wave == 7) {
        const float* cf = h_src + (size_t)n * OUT_F;
        const float* v2 = vbuf + 128;
        float a = cf[lane]      * v2[lane]
                + cf[lane + 32] * v2[lane + 32]
                + cf[lane + 64] * v2[lane + 64]
                + cf[lane + 96] * v2[lane + 96];
        for (int m = 16; m; m >>= 1) a += __shfl_xor(a, m, 32);
        if (lane == 0) s2sh = a + vbuf[256];
    }
    __syncthreads();

    const float s2 = s2sh;
    const float4 v1 = *(const float4*)(vbuf + lane * 4);        // v1[lane*4..+3]
    const float rl  = (lane < LDIM) ? rsh[lane] : 0.0f;

    // Gather neighbor rows, stash in LDS, score them.
    for (int k = wave; k < KNBR; k += 8) {
        int rowi = nbr_idx[(size_t)n * KNBR + k];
        float4 hv = *(const float4*)(h_src + (size_t)rowi * OUT_F + lane * 4);
        *(float4*)(hrows + k * OUT_F + lane * 4) = hv;

        float4 ha;
        ha.x = hv.x > 0.0f ? hv.x : 0.1f * hv.x;
        ha.y = hv.y > 0.0f ? hv.y : 0.1f * hv.y;
        ha.z = hv.z > 0.0f ? hv.z : 0.1f * hv.z;
        ha.w = hv.w > 0.0f ? hv.w : 0.1f * hv.w;
        float d  = ha.x * v1.x + ha.y * v1.y + ha.z * v1.z + ha.w * v1.w;
        float ls = (lane < LDIM) ? labels[(size_t)rowi * LDIM + lane] * rl : 0.0f;
        for (int m = 16; m; m >>= 1) {
            d  += __shfl_xor(d,  m, 32);
            ls += __shfl_xor(ls, m, 32);
        }
        if (lane == 0)
            score[k] = eps * ls + (1.0f - eps) * (d + s2);
    }
    __syncthreads();

    // softmax over K=32 (wave 0)
    if (wave == 0) {
        float v = score[lane];
        float mx = v;
        for (int m = 16; m; m >>= 1) mx = fmaxf(mx, __shfl_xor(mx, m, 32));
        float e = expf(v - mx);
        float s = e;
        for (int m = 16; m; m >>= 1) s += __shfl_xor(s, m, 32);
        score[lane] = e / s;
    }
    __syncthreads();

    // neigh = mean_k(h * att); rst = (self + neigh)*0.5 + out_bias
    if (tid < OUT_F) {
        float acc = 0.0f;
#pragma unroll
        for (int k = 0; k < KNBR; ++k)
            acc = fmaf(hrows[k * OUT_F + tid], score[k], acc);
        float neigh = acc * (1.0f / (float)KNBR);
        rst[(size_t)n * OUT_F + tid] =
            (self_out[(size_t)n * OUT_F + tid] + neigh) * 0.5f + out_bias[tid];
    }
}

// ---------------------------------------------------------------------------
// Kernel 4: passthrough outputs (cent_l, labels_ori[:n_dst])
// ---------------------------------------------------------------------------
__global__ __launch_bounds__(256) void copy_labels_kernel(
    const float* __restrict__ labels, const float* __restrict__ labels_ori,
    float* __restrict__ out_centl, float* __restrict__ out_lo, int n)
{
    int i = blockIdx.x * blockDim.x + threadIdx.x;
    if (i < n) {
        out_centl[i] = labels[i];
        out_lo[i]    = labels_ori[i];
    }
}

// ---------------------------------------------------------------------------
extern "C" void kernel_launch(void* const* d_in, const int* in_sizes, int n_in,
                              void* d_out, int out_size, void* d_ws, size_t ws_size,
                              hipStream_t stream)
{
    const float* feat       = (const float*)d_in[0];
    const float* labels     = (const float*)d_in[1];
    const float* labels_ori = (const float*)d_in[2];
    const float* label_rela = (const float*)d_in[3];
    const float* W_neigh    = (const float*)d_in[4];
    const float* b_neigh    = (const float*)d_in[5];
    const float* W_self     = (const float*)d_in[6];
    const float* b_self     = (const float*)d_in[7];
    const float* W_attn     = (const float*)d_in[8];
    const float* b_attn     = (const float*)d_in[9];
    const float* fa         = (const float*)d_in[10];
    const float* eps        = (const float*)d_in[11];
    const float* out_bias   = (const float*)d_in[12];
    const int*   nbr_idx    = (const int*)d_in[13];

    // workspace layout (floats): h_src[50000*128] | self_out[10000*128] | vbuf[257]
    float* h_src    = (float*)d_ws;
    float* self_out = h_src + (size_t)N_SRC * OUT_F;
    float* vbuf     = self_out + (size_t)N_DST * OUT_F;

    // output layout: rst[10000*128] | cent_l[10000*10] | labels_ori[10000*10]
    float* rst       = (float*)d_out;
    float* out_centl = rst + (size_t)N_DST * OUT_F;
    float* out_lo    = out_centl + (size_t)N_DST * LDIM;

    gemm_k256_n128_bias<<<(N_SRC + 255) / 256, 256, 0, stream>>>(
        feat, W_neigh, b_neigh, h_src, N_SRC);
    gemm_k256_n128_bias<<<(N_DST + 255) / 256, 256, 0, stream>>>(
        feat, W_self, b_self, self_out, N_DST);
    prep_v_kernel<<<1, 256, 0, stream>>>(W_attn, b_attn, fa, vbuf);
    copy_labels_kernel<<<(N_DST * LDIM + 255) / 256, 256, 0, stream>>>(
        labels, labels_ori, out_centl, out_lo, N_DST * LDIM);
    attn_aggregate_kernel<<<N_DST, 256, 0, stream>>>(
        h_src, self_out, labels, label_rela, vbuf, eps, out_bias, nbr_idx, rst);
}